// cartridge_49761491092114
// MI455X (gfx1250) — compile-verified
//
#include <hip/hip_runtime.h>

// Problem constants
#define IN_HW    512
#define OUT_HW   510
#define CH       3

// Tiling: 62x16 outputs per 256-thread block; input tile = 18 rows x 64 pixels
#define OUT_TW   62
#define OUT_TH   16
#define TILE_ROWS (OUT_TH + 2)          // 18 input rows
#define ROW_CHUNKS 48                   // 48 * 16B = 768B = 64 pixels * 12B
#define NCHUNK   (TILE_ROWS * ROW_CHUNKS)  // 864 b128 chunks per tile
#define LDSW     196                    // floats per LDS row (192 data + 4 pad, skews banks)

#if defined(__AMDGCN__)
typedef int v4i_t __attribute__((ext_vector_type(4)));
typedef __attribute__((address_space(1))) v4i_t g_v4i;   // global
typedef __attribute__((address_space(3))) v4i_t l_v4i;   // LDS
#endif

// ---- CDNA5 async global->LDS copy (ASYNCcnt-tracked), 16 bytes per lane ----
__device__ __forceinline__ void async_copy_b128(const void* gsrc, void* ldst) {
#if defined(__AMDGCN__)
#if __has_builtin(__builtin_amdgcn_global_load_async_to_lds_b128)
  __builtin_amdgcn_global_load_async_to_lds_b128(
      (g_v4i*)(unsigned long long)gsrc,
      (l_v4i*)(unsigned int)(unsigned long long)ldst,   // flat->LDS: low 32 bits
      0 /*imm offset*/, 0 /*cpol*/);
#else
  unsigned int l32 = (unsigned int)(unsigned long long)ldst;
  asm volatile("global_load_async_to_lds_b128 %0, %1, off"
               :: "v"(l32), "v"(gsrc) : "memory");
#endif
#else
  (void)gsrc; (void)ldst;  // host pass: never executed
#endif
}

__device__ __forceinline__ void wait_async_zero() {
#if defined(__AMDGCN__)
#if __has_builtin(__builtin_amdgcn_s_wait_asynccnt)
  __builtin_amdgcn_s_wait_asynccnt(0);
#else
  asm volatile("s_wait_asynccnt 0" ::: "memory");
#endif
#endif
}

__global__ __launch_bounds__(256) void cartridge_49761491092114_kernel(
    const float* __restrict__ x, const float* __restrict__ kern,
    float* __restrict__ out)
{
  __shared__ float tile[TILE_ROWS * LDSW];

  const int tid = threadIdx.x;
  const int b   = blockIdx.z;
  int h0 = blockIdx.y * OUT_TH; if (h0 > OUT_HW - OUT_TH) h0 = OUT_HW - OUT_TH; // <= 494
  int w0 = blockIdx.x * OUT_TW; if (w0 > OUT_HW - OUT_TW) w0 = OUT_HW - OUT_TW; // <= 448

  const size_t rowF = (size_t)IN_HW * CH;                       // 1536 floats per input row
  const float* gbase = x + ((size_t)b * IN_HW + h0) * rowF + (size_t)w0 * CH;

  // ---- Stage the 18 x 768B input tile into LDS via async b128 loads ----
#pragma unroll
  for (int it = 0; it < 4; ++it) {
    int k = tid + it * 256;
    if (k < NCHUNK) {
      int i = k / ROW_CHUNKS;           // tile row 0..17
      int j = k - i * ROW_CHUNKS;       // 16B chunk 0..47 (stays inside the image row)
      async_copy_b128(gbase + (size_t)i * rowF + j * 4,
                      &tile[i * LDSW + j * 4]);
    }
  }
  wait_async_zero();
  __syncthreads();

  // Weight groups: taps (0,0),(1,0),(0,1) use (k0,k2,k4); taps (2,1),(1,2),(2,2) use (k1,k3,k5)
  const float wa0 = kern[0], wb0 = kern[1];
  const float wa1 = kern[2], wb1 = kern[3];
  const float wa2 = kern[4], wb2 = kern[5];

  // ---- Each thread: 4 consecutive output rows of one output column ----
  const int tc = tid & 63;              // output column in tile (wave-contiguous)
  const int tr = tid >> 6;              // 0..3
  if (tc < OUT_TW) {
    const int r0 = tr * 4;

    // a(i,j) = pixel . (wa0,wa1,wa2) ; b(i,j) = pixel . (wb0,wb1,wb2)
    // out(r,c) = a(r,c) + a(r+1,c) + a(r,c+1) + b(r+2,c+1) + b(r+1,c+2) + b(r+2,c+2)
    float a0[5], a1[4], b1[4], b2[5];
#pragma unroll
    for (int i = 0; i < 5; ++i) {       // column c, rows r0..r0+4 -> a
      const float* p = &tile[(r0 + i) * LDSW + tc * CH];
      a0[i] = p[0] * wa0 + p[1] * wa1 + p[2] * wa2;
    }
#pragma unroll
    for (int i = 0; i < 6; ++i) {       // column c+1, rows r0..r0+5 -> a (first 4), b (last 4)
      const float* p = &tile[(r0 + i) * LDSW + (tc + 1) * CH];
      float e0 = p[0], e1 = p[1], e2 = p[2];
      if (i < 4)  a1[i]     = e0 * wa0 + e1 * wa1 + e2 * wa2;
      if (i >= 2) b1[i - 2] = e0 * wb0 + e1 * wb1 + e2 * wb2;
    }
#pragma unroll
    for (int i = 0; i < 5; ++i) {       // column c+2, rows r0+1..r0+5 -> b
      const float* p = &tile[(r0 + 1 + i) * LDSW + (tc + 2) * CH];
      b2[i] = p[0] * wb0 + p[1] * wb1 + p[2] * wb2;
    }

    float* obase = out + ((size_t)b * OUT_HW + (h0 + r0)) * OUT_HW + (w0 + tc);
#pragma unroll
    for (int r = 0; r < 4; ++r) {
      float v = a0[r] + a0[r + 1] + a1[r] + b1[r] + b2[r] + b2[r + 1];
      __builtin_nontemporal_store(v, obase + (size_t)r * OUT_HW); // write-once stream
    }
  }
}

extern "C" void kernel_launch(void* const* d_in, const int* in_sizes, int n_in,
                              void* d_out, int out_size, void* d_ws, size_t ws_size,
                              hipStream_t stream) {
  (void)in_sizes; (void)n_in; (void)out_size; (void)d_ws; (void)ws_size;
  const float* x    = (const float*)d_in[0];   // (128, 512, 512, 3) f32
  const float* kern = (const float*)d_in[1];   // (6, 1) f32
  float* out = (float*)d_out;                  // (128, 510, 510, 1) f32

  dim3 grid((OUT_HW + OUT_TW - 1) / OUT_TW,    // 9
            (OUT_HW + OUT_TH - 1) / OUT_TH,    // 32
            128);
  cartridge_49761491092114_kernel<<<grid, dim3(256), 0, stream>>>(x, kern, out);
}